// Encoder_71313636983306
// MI455X (gfx1250) — compile-verified
//
#include <hip/hip_runtime.h>

typedef __attribute__((ext_vector_type(2))) float v2f;
typedef __attribute__((ext_vector_type(8))) float v8f;

#define NPTS   100000
#define NBATCH 2
#define NCH    64
#define PGRID  (640 * 640)
#define BN_EPS 1e-5f

// Fold BatchNorm (eval) into the linear layer, with homogeneous 4th column:
// Wf[d][0..2] = scale_d * W[d][0..2];  Wf[d][3] = scale_d*(b_d - mean_d) + beta_d
__global__ void fold_params_kernel(const float* __restrict__ W,
                                   const float* __restrict__ b,
                                   const float* __restrict__ gamma,
                                   const float* __restrict__ beta,
                                   const float* __restrict__ mean,
                                   const float* __restrict__ var,
                                   float* __restrict__ Wf) {
    int d = threadIdx.x;
    if (d < NCH) {
        float s = gamma[d] * rsqrtf(var[d] + BN_EPS);
        Wf[d * 4 + 0] = s * W[d * 3 + 0];
        Wf[d * 4 + 1] = s * W[d * 3 + 1];
        Wf[d * 4 + 2] = s * W[d * 3 + 2];
        Wf[d * 4 + 3] = s * (b[d] - mean[d]) + beta[d];
    }
}

// Zero the 419MB output with 128-bit stores (harness poisons d_out).
__global__ void zero_out_kernel(float4* __restrict__ out, int n4) {
    int i = blockIdx.x * blockDim.x + threadIdx.x;
    if (i < n4) out[i] = make_float4(0.f, 0.f, 0.f, 0.f);
}

// One wave32 per 16-point tile. A = [p | 1] (16x4), B = Wf^T slice (4x16),
// D = 16 points x 16 channels of the embedding, 4 WMMAs cover 64 channels.
// ReLU + mask applied at scatter time; zeros skip the atomic entirely.
__global__ __launch_bounds__(256)
void embed_scatter_kernel(const float* __restrict__ pcl,           // [B*N, 3]
                          const unsigned char* __restrict__ mask,  // [B*N]
                          const int* __restrict__ gridIdx,         // [B*N]
                          const float* __restrict__ Wf,            // [64, 4]
                          float* __restrict__ out,                 // [4, 64, P]
                          int s,                                   // 0=prev, 1=cur
                          int ntiles) {
    int lane = threadIdx.x & 31;
    int tile = (blockIdx.x * blockDim.x + threadIdx.x) >> 5;  // wave-uniform
    if (tile >= ntiles) return;                               // whole wave exits

    const int half = lane >> 4;   // 0: K=0,1   1: K=2,3
    const int nn   = lane & 15;   // point (for A) / channel-in-group (for B,D)
    const int base = tile * 16;   // flat point index base within [B*N]

    // ---- A matrix: homogeneous point coords ----
    int f = base + nn;
    v2f a;
    if (half == 0) {
        a.x = pcl[f * 3 + 0];
        a.y = pcl[f * 3 + 1];
    } else {
        a.x = pcl[f * 3 + 2];
        a.y = 1.0f;
    }

    // ---- per-lane scatter metadata: D VGPR r holds point m = r + 8*half ----
    const int mbase = base + 8 * half;
    int idxv[8];
    unsigned char mk[8];
#pragma unroll
    for (int r = 0; r < 8; ++r) {
        idxv[r] = gridIdx[mbase + r];
        mk[r]   = mask[mbase + r];
    }

    // N=100000 is a multiple of 16, so the whole tile shares one batch index.
    const int bidx = base / NPTS;
    float* outBase = out + (size_t)(bidx * 2 + s) * NCH * (size_t)PGRID;

#pragma unroll
    for (int g = 0; g < 4; ++g) {
        const int ch = g * 16 + nn;  // output channel for this lane
        v2f bm;
        bm.x = Wf[ch * 4 + 2 * half + 0];  // B[k][n] = Wf[n][k], k = 2*half
        bm.y = Wf[ch * 4 + 2 * half + 1];

        v8f c = {};
        c = __builtin_amdgcn_wmma_f32_16x16x4_f32(
                /*neg_a=*/false, a, /*neg_b=*/false, bm,
                /*c_mod=*/(short)0, c, /*reuse_a=*/false, /*reuse_b=*/false);

#pragma unroll
        for (int r = 0; r < 8; ++r) {
            float h = c[r];                 // point m=r+8*half, channel ch
            if (mk[r] && h > 0.0f) {        // ReLU + mask; skip no-op atomics
                atomicAdd(outBase + (size_t)ch * PGRID + idxv[r], h);
            }
        }
    }
}

extern "C" void kernel_launch(void* const* d_in, const int* in_sizes, int n_in,
                              void* d_out, int out_size, void* d_ws, size_t ws_size,
                              hipStream_t stream) {
    const float*         prev_pcl  = (const float*)d_in[0];
    const unsigned char* prev_mask = (const unsigned char*)d_in[1];
    const int*           prev_grid = (const int*)d_in[2];
    const float*         cur_pcl   = (const float*)d_in[3];
    const unsigned char* cur_mask  = (const unsigned char*)d_in[4];
    const int*           cur_grid  = (const int*)d_in[5];
    const float* W     = (const float*)d_in[6];
    const float* b     = (const float*)d_in[7];
    const float* gamma = (const float*)d_in[8];
    const float* beta  = (const float*)d_in[9];
    const float* mean  = (const float*)d_in[10];
    const float* var   = (const float*)d_in[11];

    float* out = (float*)d_out;
    float* Wf  = (float*)d_ws;  // 64*4 floats = 1KB

    fold_params_kernel<<<1, 64, 0, stream>>>(W, b, gamma, beta, mean, var, Wf);

    int n4 = out_size / 4;  // 104,857,600 / 4 = 26,214,400 float4 stores
    zero_out_kernel<<<(n4 + 255) / 256, 256, 0, stream>>>((float4*)d_out, n4);

    const int ntiles = (NBATCH * NPTS) / 16;          // 12500 tiles per cloud
    const int blocks = (ntiles * 32 + 255) / 256;     // 1 wave per tile
    embed_scatter_kernel<<<blocks, 256, 0, stream>>>(prev_pcl, prev_mask, prev_grid,
                                                     Wf, out, 0, ntiles);
    embed_scatter_kernel<<<blocks, 256, 0, stream>>>(cur_pcl, cur_mask, cur_grid,
                                                     Wf, out, 1, ntiles);
}